// VoxelModel_31774168055891
// MI455X (gfx1250) — compile-verified
//
#include <hip/hip_runtime.h>
#include <cmath>

// Volume-rendering composite (DVGO-style Raw2Alpha + Alphas2Weights + white-bg
// marching) for MI455X / gfx1250.
//
// Roofline: 20 B/sample * 4.19M samples ~= 84 MB @ 23.3 TB/s => ~3.6 us floor.
// INTERVAL = 0.5 gives om = exp(-0.5*softplus(x)) = rsqrt(1 + e^x) and the
// cancellation-free alpha = u*om / (1 + (1+u)*om) == 1 - om  (u = e^x), so the
// transcendental budget is v_exp/v_rsq/v_rcp only. Each lane handles 4
// consecutive samples (128/chunk/wave) so density & rgb stream as b128 loads
// and the wave product-scan + loop overhead amortize 4x, keeping instruction
// issue (~1.6 us) well under the HBM time. No WMMA: no matmul structure.

__global__ __launch_bounds__(256) void vm_init_bounds(int* __restrict__ ray_start, int N) {
    int r = blockIdx.x * blockDim.x + threadIdx.x;
    if (r < N) ray_start[r] = -1;
}

__global__ __launch_bounds__(256) void vm_find_bounds(const int* __restrict__ ray_id, int M,
                                                      int* __restrict__ ray_start,
                                                      int* __restrict__ ray_end) {
    int i = blockIdx.x * blockDim.x + threadIdx.x;
    if (i >= M) return;
    int r = ray_id[i];
    if (i == 0 || ray_id[i - 1] != r) ray_start[r] = i;          // first sample of segment
    if (i == M - 1 || ray_id[i + 1] != r) ray_end[r] = i + 1;    // one past last sample
}

__device__ __forceinline__ float fast_rcp(float x) { return __builtin_amdgcn_rcpf(x); }
__device__ __forceinline__ float fast_rsq(float x) { return __builtin_amdgcn_rsqf(x); }

__device__ __forceinline__ float sigmoid_fast(float x) {
    return fast_rcp(1.0f + __expf(-x));   // overflow -> inf -> rcp -> 0: safe
}

// One wave (32 lanes) per ray; 4 consecutive samples per lane per chunk.
__global__ __launch_bounds__(256) void vm_march(const float* __restrict__ density,
                                                const float* __restrict__ rgb_raw,
                                                const float* __restrict__ shift,
                                                const int* __restrict__ ray_start,
                                                const int* __restrict__ ray_end,
                                                float* __restrict__ out,
                                                int N) {
    const int lane = threadIdx.x & 31;
    const int wave = threadIdx.x >> 5;
    const int r = blockIdx.x * 8 + wave;
    if (r >= N) return;

    const float sh = shift[0];

    float accR = 0.0f, accG = 0.0f, accB = 0.0f;
    float carry = 1.0f;                          // running per-ray transmittance

    const int s0 = ray_start[r];
    if (s0 >= 0) {
        const int e = ray_end[r];
        const int base0 = s0 & ~3;               // 16B-align the sample grid

        for (int base = base0; base < e; base += 128) {
            const int i0 = base + 4 * lane;      // this lane's first sample

            // Stream-ahead prefetch of the next chunk (global_prefetch_b8).
            {
                int pf = i0 + 128;
                if (pf < e) {
                    __builtin_prefetch(&density[pf], 0, 3);
                    __builtin_prefetch(&rgb_raw[3 * (size_t)pf], 0, 3);
                }
            }

            const bool full_ld = (i0 + 3 < e);   // whole float4 in-bounds

            // ---- density: one b128 load on the hot path -------------------
            float d[4];
            if (full_ld) {
                const float4 v = *reinterpret_cast<const float4*>(density + i0);
                d[0] = v.x; d[1] = v.y; d[2] = v.z; d[3] = v.w;
            } else {
                #pragma unroll
                for (int j = 0; j < 4; ++j)
                    d[j] = (i0 + j < e) ? density[i0 + j] : 0.0f;
            }

            // ---- rgb: three b128 loads on the hot path --------------------
            float c[12];
            if (full_ld) {
                const float4* cp = reinterpret_cast<const float4*>(rgb_raw + 3 * (size_t)i0);
                float4 v0 = cp[0], v1 = cp[1], v2 = cp[2];
                c[0] = v0.x; c[1]  = v0.y; c[2]  = v0.z; c[3]  = v0.w;
                c[4] = v1.x; c[5]  = v1.y; c[6]  = v1.z; c[7]  = v1.w;
                c[8] = v2.x; c[9]  = v2.y; c[10] = v2.z; c[11] = v2.w;
            } else {
                #pragma unroll
                for (int j = 0; j < 4; ++j) {
                    if (i0 + j < e) {
                        const float* p = rgb_raw + 3 * (size_t)(i0 + j);
                        c[3*j] = p[0]; c[3*j+1] = p[1]; c[3*j+2] = p[2];
                    } else {
                        c[3*j] = 0.0f; c[3*j+1] = 0.0f; c[3*j+2] = 0.0f;
                    }
                }
            }

            // ---- per-sample alpha / (1-alpha), masked outside [s0, e) -----
            float om[4], alpha[4];
            #pragma unroll
            for (int j = 0; j < 4; ++j) {
                const int  i = i0 + j;
                const bool a = (i >= s0) && (i < e);
                float x  = fminf(d[j] + sh, 80.0f);
                float u  = a ? __expf(x) : 0.0f;   // inactive -> om=1, alpha=0
                float w1 = 1.0f + u;
                float o  = fast_rsq(w1);           // (1+u)^(-1/2)
                float t  = w1 * o;                 // sqrt(1+u)
                om[j]    = o;
                alpha[j] = u * o * fast_rcp(1.0f + t);   // == 1 - om, stable
            }

            // ---- wave scan of 4-sample products (one scan / 128 samples) --
            float p = om[0] * om[1] * om[2] * om[3];
            #pragma unroll
            for (int off = 1; off < 32; off <<= 1) {
                float s = __shfl_up(p, off, 32);
                if (lane >= off) p *= s;
            }
            float excl = __shfl_up(p, 1, 32);
            if (lane == 0) excl = 1.0f;

            // ---- composit 4 samples serially in-lane ----------------------
            float T = carry * excl;
            #pragma unroll
            for (int j = 0; j < 4; ++j) {
                const float w = alpha[j] * T;
                accR += w * sigmoid_fast(c[3*j]);
                accG += w * sigmoid_fast(c[3*j+1]);
                accB += w * sigmoid_fast(c[3*j+2]);
                T *= om[j];
            }

            carry *= __shfl(p, 31, 32);            // uniform across lanes
        }
    }

    // Wave reduction of the color accumulators.
    #pragma unroll
    for (int off = 16; off > 0; off >>= 1) {
        accR += __shfl_down(accR, off, 32);
        accG += __shfl_down(accG, off, 32);
        accB += __shfl_down(accB, off, 32);
    }

    if (lane == 0) {
        // White background: + leftover transmittance (== 1 for empty rays).
        out[3 * (size_t)r + 0] = accR + carry;
        out[3 * (size_t)r + 1] = accG + carry;
        out[3 * (size_t)r + 2] = accB + carry;
    }
}

extern "C" void kernel_launch(void* const* d_in, const int* in_sizes, int n_in,
                              void* d_out, int out_size, void* d_ws, size_t ws_size,
                              hipStream_t stream) {
    const float* density = (const float*)d_in[0];
    const float* rgb_raw = (const float*)d_in[1];
    const float* shift   = (const float*)d_in[2];
    const int*   ray_id  = (const int*)d_in[3];
    // d_in[4] is N on device; host-side launch config derives it from out_size.
    const int M = in_sizes[0];
    const int N = out_size / 3;

    int* ray_start = (int*)d_ws;         // N ints
    int* ray_end   = ray_start + N;      // N ints
    (void)ws_size; (void)n_in;

    vm_init_bounds<<<(N + 255) / 256, 256, 0, stream>>>(ray_start, N);
    vm_find_bounds<<<(M + 255) / 256, 256, 0, stream>>>(ray_id, M, ray_start, ray_end);

    const int blocks = (N + 7) / 8;      // 8 rays (waves) per 256-thread block
    vm_march<<<blocks, 256, 0, stream>>>(density, rgb_raw, shift,
                                         ray_start, ray_end,
                                         (float*)d_out, N);
}